// LRUUnet_61057255080147
// MI455X (gfx1250) — compile-verified
//
#include <hip/hip_runtime.h>
#include <hip/hip_bf16.h>
#include <math.h>

// ---------------------------------------------------------------------------
// CDNA5 (gfx1250) implementation of the LRU U-Net forward pass.
// fp32 end-to-end; all dense projections use V_WMMA_F32_16X16X4_F32.
// Linear recurrence parallelized with a 3-pass chunked scan (chunk = 256).
// ---------------------------------------------------------------------------

typedef __attribute__((ext_vector_type(2))) float v2f;
typedef __attribute__((ext_vector_type(8))) float v8f;

static inline int cdiv(int a, int b) { return (a + b - 1) / b; }

// ---------------------------------------------------------------------------
// GEMM: Y[M x NN] = A[M x K] * W[K x NN] (+ bias). One wave per 16-row stripe.
// f32 WMMA 16x16x4. A-frag: lane(0-15)=row, VGPRs hold k0+2*hi .. +1.
// B-frag: lane(0-15)=col, VGPRs hold rows k0+2*hi .. +1.
// ---------------------------------------------------------------------------
template <int K, int NN, bool BIAS>
__global__ __launch_bounds__(256) void gemm_wmma(
    const float* __restrict__ A, const float* __restrict__ W,
    const float* __restrict__ bias, float* __restrict__ Y, int M) {
  const int wave = (blockIdx.x * 256 + (int)threadIdx.x) >> 5;
  const int lane = threadIdx.x & 31;
  const int row0 = wave * 16;
  if (row0 >= M) return;
  const int lr = lane & 15;
  const int hi = lane >> 4;
  constexpr int NT = NN / 16;
  v8f acc[NT];
#pragma unroll
  for (int j = 0; j < NT; ++j) acc[j] = (v8f)(0.0f);
  const float* __restrict__ arow = A + (size_t)(row0 + lr) * K + 2 * hi;
  for (int k0 = 0; k0 < K; k0 += 4) {
    v2f a;
    a.x = arow[k0];
    a.y = arow[k0 + 1];
#pragma unroll
    for (int j = 0; j < NT; ++j) {
      const int col = j * 16 + lr;
      v2f b;
      b.x = W[(size_t)(k0 + 2 * hi) * NN + col];
      b.y = W[(size_t)(k0 + 2 * hi + 1) * NN + col];
      acc[j] = __builtin_amdgcn_wmma_f32_16x16x4_f32(
          false, a, false, b, (short)0, acc[j], false, false);
    }
  }
#pragma unroll
  for (int j = 0; j < NT; ++j) {
    const int col = j * 16 + lr;
    const float bv = BIAS ? bias[col] : 0.0f;
#pragma unroll
    for (int r = 0; r < 8; ++r) {
      const int row = row0 + r + 8 * hi;
      Y[(size_t)row * NN + col] = acc[j][r] + bv;
    }
  }
}

// ---------------------------------------------------------------------------
// LRU output projection: Y = A[M x K]*W[K x H] + D.*U, then optional LN+SiLU.
// C/D tile layout: VGPR r holds row (row0 + r + 8*hi), col = j*16 + (lane&15),
// so a full matrix row lives in 16 lanes of one half-wave -> shfl_xor 1/2/4/8.
// ---------------------------------------------------------------------------
template <int K, int H, bool ACT>
__global__ __launch_bounds__(256) void gemm_lru_out(
    const float* __restrict__ A, const float* __restrict__ W,
    const float* __restrict__ U, const float* __restrict__ D,
    const float* __restrict__ lnw, const float* __restrict__ lnb,
    float* __restrict__ Y, int M) {
  const int wave = (blockIdx.x * 256 + (int)threadIdx.x) >> 5;
  const int lane = threadIdx.x & 31;
  const int row0 = wave * 16;
  if (row0 >= M) return;
  const int lr = lane & 15;
  const int hi = lane >> 4;
  constexpr int NT = H / 16;
  v8f acc[NT];
#pragma unroll
  for (int j = 0; j < NT; ++j) acc[j] = (v8f)(0.0f);
  const float* __restrict__ arow = A + (size_t)(row0 + lr) * K + 2 * hi;
  for (int k0 = 0; k0 < K; k0 += 4) {
    v2f a;
    a.x = arow[k0];
    a.y = arow[k0 + 1];
#pragma unroll
    for (int j = 0; j < NT; ++j) {
      const int col = j * 16 + lr;
      v2f b;
      b.x = W[(size_t)(k0 + 2 * hi) * H + col];
      b.y = W[(size_t)(k0 + 2 * hi + 1) * H + col];
      acc[j] = __builtin_amdgcn_wmma_f32_16x16x4_f32(
          false, a, false, b, (short)0, acc[j], false, false);
    }
  }
  float val[NT][8];
#pragma unroll
  for (int j = 0; j < NT; ++j) {
    const int col = j * 16 + lr;
    const float d = D[col];
#pragma unroll
    for (int r = 0; r < 8; ++r) {
      const int row = row0 + r + 8 * hi;
      val[j][r] = acc[j][r] + d * U[(size_t)row * H + col];
    }
  }
  if (ACT) {
#pragma unroll
    for (int r = 0; r < 8; ++r) {
      float s = 0.0f, q = 0.0f;
#pragma unroll
      for (int j = 0; j < NT; ++j) {
        const float v = val[j][r];
        s += v;
        q += v * v;
      }
#pragma unroll
      for (int m = 8; m >= 1; m >>= 1) {
        s += __shfl_xor(s, m, 32);
        q += __shfl_xor(q, m, 32);
      }
      const float mean = s * (1.0f / H);
      const float var = q * (1.0f / H) - mean * mean;
      const float rstd = rsqrtf(var + 1e-5f);
#pragma unroll
      for (int j = 0; j < NT; ++j) {
        const int col = j * 16 + lr;
        const float g = (val[j][r] - mean) * rstd * lnw[col] + lnb[col];
        val[j][r] = g / (1.0f + expf(-g));  // SiLU
      }
    }
  }
#pragma unroll
  for (int j = 0; j < NT; ++j) {
    const int col = j * 16 + lr;
#pragma unroll
    for (int r = 0; r < 8; ++r) {
      const int row = row0 + r + 8 * hi;
      Y[(size_t)row * H + col] = val[j][r];
    }
  }
}

// ---------------------------------------------------------------------------
// LRU parameter prep: lambda = exp(-exp(nu) + i*exp(theta)); lamP = lambda^256;
// Wb[h, n | N+n] = (B_re|B_im)[n,h]*exp(gamma[n])   (K=H, NN=2N GEMM weight)
// Wc[n | N+n, h] = ( C_re | -C_im )[h,n]            (K=2N, NN=H GEMM weight)
// ---------------------------------------------------------------------------
template <int H>
__global__ void prep_lru(const float* __restrict__ nu, const float* __restrict__ th,
                         const float* __restrict__ ga, const float* __restrict__ Bre,
                         const float* __restrict__ Bim, const float* __restrict__ Cre,
                         const float* __restrict__ Cim, float* lam_re, float* lam_im,
                         float* lamPr, float* lamPi, float* Wb, float* Wc) {
  constexpr int N = H, N2 = 2 * H;
  const int tid = threadIdx.x;
  if (tid < N) {
    const float mod = expf(-expf(nu[tid]));
    const float ang = expf(th[tid]);
    const float lrr = mod * cosf(ang), lii = mod * sinf(ang);
    lam_re[tid] = lrr;
    lam_im[tid] = lii;
    float pr = lrr, pi = lii;  // lambda^(2^8) via 8 squarings
    for (int s = 0; s < 8; ++s) {
      const float nr = pr * pr - pi * pi;
      const float ni = 2.0f * pr * pi;
      pr = nr;
      pi = ni;
    }
    lamPr[tid] = pr;
    lamPi[tid] = pi;
  }
  for (int idx = tid; idx < H * N2; idx += blockDim.x) {
    const int h = idx / N2, col = idx % N2;
    const int n = (col < N) ? col : (col - N);
    const float g = expf(ga[n]);
    Wb[idx] = ((col < N) ? Bre[n * H + h] : Bim[n * H + h]) * g;
  }
  for (int idx = tid; idx < N2 * H; idx += blockDim.x) {
    const int k = idx / H, h = idx % H;
    const int n = (k < N) ? k : (k - N);
    Wc[idx] = (k < N) ? Cre[h * N + n] : -Cim[h * N + n];
  }
}

// ---------------------------------------------------------------------------
// Chunked complex linear scan: x_t = lam*x_{t-1} + bu_t, chunk = 256 steps.
// bu / xs layout: (B, T, 2N) with re in cols [0,N), im in cols [N,2N).
// ---------------------------------------------------------------------------
__global__ void lru_scan_pass1(const float* __restrict__ bu,
                               const float* __restrict__ lam_re,
                               const float* __restrict__ lam_im,
                               float* __restrict__ carry, int N, int T, int NC,
                               int total) {
  const int tid = blockIdx.x * blockDim.x + threadIdx.x;
  if (tid >= total) return;
  const int n = tid % N;
  const int c = (tid / N) % NC;
  const int b = tid / (N * NC);
  const float lrr = lam_re[n], lii = lam_im[n];
  const float* __restrict__ p = bu + ((size_t)b * T + (size_t)c * 256) * (2 * N);
  float sr = 0.0f, si = 0.0f;
  for (int t = 0; t < 256; ++t) {
    const float br = p[(size_t)t * 2 * N + n];
    const float bi = p[(size_t)t * 2 * N + N + n];
    const float nr = fmaf(sr, lrr, fmaf(-si, lii, br));
    const float ni = fmaf(sr, lii, fmaf(si, lrr, bi));
    sr = nr;
    si = ni;
  }
  float* q = carry + ((size_t)b * NC + c) * (2 * N);
  q[n] = sr;
  q[N + n] = si;
}

__global__ void lru_scan_pass2(const float* __restrict__ carry,
                               const float* __restrict__ lamPr,
                               const float* __restrict__ lamPi,
                               float* __restrict__ inits, int N, int NC, int total) {
  const int tid = blockIdx.x * blockDim.x + threadIdx.x;
  if (tid >= total) return;
  const int n = tid % N;
  const int b = tid / N;
  const float pr = lamPr[n], pi = lamPi[n];
  float sr = 0.0f, si = 0.0f;
  for (int c = 0; c < NC; ++c) {
    const size_t off = ((size_t)b * NC + c) * (2 * N);
    inits[off + n] = sr;          // state entering chunk c
    inits[off + N + n] = si;
    const float cr = carry[off + n];
    const float ci = carry[off + N + n];
    const float nr = fmaf(sr, pr, fmaf(-si, pi, cr));
    const float ni = fmaf(sr, pi, fmaf(si, pr, ci));
    sr = nr;
    si = ni;
  }
}

__global__ void lru_scan_pass3(const float* __restrict__ bu,
                               const float* __restrict__ lam_re,
                               const float* __restrict__ lam_im,
                               const float* __restrict__ inits,
                               float* __restrict__ xs, int N, int T, int NC,
                               int total) {
  const int tid = blockIdx.x * blockDim.x + threadIdx.x;
  if (tid >= total) return;
  const int n = tid % N;
  const int c = (tid / N) % NC;
  const int b = tid / (N * NC);
  const float lrr = lam_re[n], lii = lam_im[n];
  const size_t coff = ((size_t)b * NC + c) * (2 * N);
  float sr = inits[coff + n], si = inits[coff + N + n];
  const size_t base = ((size_t)b * T + (size_t)c * 256) * (2 * N);
  const float* __restrict__ p = bu + base;
  float* __restrict__ o = xs + base;
  for (int t = 0; t < 256; ++t) {
    const float br = p[(size_t)t * 2 * N + n];
    const float bi = p[(size_t)t * 2 * N + N + n];
    const float nr = fmaf(sr, lrr, fmaf(-si, lii, br));
    const float ni = fmaf(sr, lii, fmaf(si, lrr, bi));
    sr = nr;
    si = ni;
    o[(size_t)t * 2 * N + n] = sr;
    o[(size_t)t * 2 * N + N + n] = si;
  }
}

// ---------------------------------------------------------------------------
// Conv weight repacks (stride-4 conv <-> GEMM over contiguous 4-step rows).
// ---------------------------------------------------------------------------
__global__ void prep_down_w(const float* __restrict__ w, float* __restrict__ Wd,
                            int Cin, int Cout) {
  // Wd[(j*Cin + c)*Cout + o] = w[(o*Cin + c)*4 + j]
  const int total = 4 * Cin * Cout;
  for (int idx = blockIdx.x * blockDim.x + threadIdx.x; idx < total;
       idx += gridDim.x * blockDim.x) {
    const int o = idx % Cout;
    const int rc = idx / Cout;
    const int c = rc % Cin;
    const int j = rc / Cin;
    Wd[idx] = w[(o * Cin + c) * 4 + j];
  }
}

__global__ void prep_up_w(const float* __restrict__ w, float* __restrict__ Wu,
                          int Cin, int Cout) {
  // Wu[c*(4*Cout) + j*Cout + o] = w[(c*Cout + o)*4 + j]
  const int total = 4 * Cin * Cout;
  for (int idx = blockIdx.x * blockDim.x + threadIdx.x; idx < total;
       idx += gridDim.x * blockDim.x) {
    const int o = idx % Cout;
    const int rj = idx / Cout;
    const int j = rj % 4;
    const int c = rj / 4;
    Wu[idx] = w[(c * Cout + o) * 4 + j];
  }
}

// up-conv epilogue: causal shift by one step, bias, skip add
__global__ void upfix(const float* __restrict__ yraw, const float* __restrict__ bias,
                      const float* __restrict__ skip, float* __restrict__ out,
                      int T, int C, int total) {
  const int idx = blockIdx.x * blockDim.x + threadIdx.x;
  if (idx >= total) return;
  const int c = idx % C;
  const int t = (idx / C) % T;
  const float v = (t > 0) ? (yraw[idx - C] + bias[c]) : 0.0f;
  out[idx] = v + skip[idx];
}

// (B,C,T) -> (B,T,C)
__global__ void tr_ct_tc(const float* __restrict__ x, float* __restrict__ h,
                         int C, int T, int total) {
  const int idx = blockIdx.x * blockDim.x + threadIdx.x;
  if (idx >= total) return;
  const int c = idx % C;
  const int t = (idx / C) % T;
  const int b = idx / (C * T);
  h[idx] = x[((size_t)b * C + c) * T + t];
}

// (B,T,C) -> (B,C,T)
__global__ void tr_tc_ct(const float* __restrict__ h, float* __restrict__ y,
                         int C, int T, int total) {
  const int idx = blockIdx.x * blockDim.x + threadIdx.x;
  if (idx >= total) return;
  const int t = idx % T;
  const int c = (idx / T) % C;
  const int b = idx / (C * T);
  y[idx] = h[((size_t)b * T + t) * C + c];
}

// ---------------------------------------------------------------------------
// Host orchestration
// ---------------------------------------------------------------------------
namespace {

struct LruP {
  const float *nu, *th, *ga, *Bre, *Bim, *Cre, *Cim, *D, *lnw, *lnb;
};

struct Ws {
  float *actA, *actB, *actC, *skip1;
  float *bu, *xs;
  float *lam_re, *lam_im, *lamPr, *lamPi, *Wb, *Wc, *Wconv;
  float *carry, *inits;
};

template <int H, bool ACT>
void lru_block(const LruP& p, const float* u, float* y, const Ws& w, int BT,
               hipStream_t s) {
  constexpr int N2 = 2 * H;
  prep_lru<H><<<1, 256, 0, s>>>(p.nu, p.th, p.ga, p.Bre, p.Bim, p.Cre, p.Cim,
                                w.lam_re, w.lam_im, w.lamPr, w.lamPi, w.Wb, w.Wc);
  gemm_wmma<H, N2, false><<<BT / 128, 256, 0, s>>>(u, w.Wb, nullptr, w.bu, BT);
  const int T = BT / 8;             // batch = 8
  const int NC = T / 256;           // chunk = 256
  const int tot1 = 8 * NC * H;
  const int tot2 = 8 * H;
  lru_scan_pass1<<<cdiv(tot1, 256), 256, 0, s>>>(w.bu, w.lam_re, w.lam_im,
                                                 w.carry, H, T, NC, tot1);
  lru_scan_pass2<<<cdiv(tot2, 256), 256, 0, s>>>(w.carry, w.lamPr, w.lamPi,
                                                 w.inits, H, NC, tot2);
  lru_scan_pass3<<<cdiv(tot1, 256), 256, 0, s>>>(w.bu, w.lam_re, w.lam_im,
                                                 w.inits, w.xs, H, T, NC, tot1);
  gemm_lru_out<N2, H, ACT><<<BT / 128, 256, 0, s>>>(w.xs, w.Wc, u, p.D, p.lnw,
                                                    p.lnb, y, BT);
}

}  // namespace

extern "C" void kernel_launch(void* const* d_in, const int* in_sizes, int n_in,
                              void* d_out, int out_size, void* d_ws, size_t ws_size,
                              hipStream_t stream) {
  (void)in_sizes; (void)n_in; (void)out_size; (void)ws_size;
  constexpr int B = 8;
  constexpr int T0 = 32768;
  constexpr int BT = B * T0;  // 262144

  // ---- unpack inputs (depth-first insertion order of setup_inputs dict) ----
  int i = 0;
  auto nxt = [&]() { return (const float*)d_in[i++]; };
  const float* x = nxt();  // (B, 16, T0)
  auto get_block = [&](LruP& p) {
    p.nu = nxt(); p.th = nxt(); p.ga = nxt();
    p.Bre = nxt(); p.Bim = nxt(); p.Cre = nxt(); p.Cim = nxt(); p.D = nxt();
    p.lnw = nxt(); p.lnb = nxt();
  };
  LruP d0; get_block(d0); const float* cw0 = nxt(); const float* cb0 = nxt();
  LruP d1; get_block(d1); const float* cw1 = nxt(); const float* cb1 = nxt();
  LruP h0; get_block(h0);
  LruP h1; get_block(h1);
  const float* uw0 = nxt(); const float* ub0 = nxt(); LruP u0; get_block(u0);
  const float* uw1 = nxt(); const float* ub1 = nxt(); LruP u1; get_block(u1);
  LruP l0; get_block(l0);
  LruP l1; get_block(l1);

  // ---- workspace layout (floats) ----
  const size_t F1M = 1u << 20;
  float* base = (float*)d_ws;
  size_t o = 0;
  Ws w;
  w.actA  = base + o; o += 4 * F1M;   // (B, T0, 16)
  w.actB  = base + o; o += 4 * F1M;
  w.actC  = base + o; o += 2 * F1M;
  w.skip1 = base + o; o += 2 * F1M;   // (B, T0/4, 32)
  w.bu    = base + o; o += 8 * F1M;   // (B, T, 2N) max
  w.xs    = base + o; o += 8 * F1M;
  w.lam_re = base + o; o += 64;
  w.lam_im = base + o; o += 64;
  w.lamPr  = base + o; o += 64;
  w.lamPi  = base + o; o += 64;
  w.Wb    = base + o; o += 8192;
  w.Wc    = base + o; o += 8192;
  w.Wconv = base + o; o += 8192;
  w.carry = base + o; o += 65536;
  w.inits = base + o; o += 65536;

  const int totTC = B * 16 * T0;

  // 1) (B,16,T) -> (B,T,16); actA doubles as skip0
  tr_ct_tc<<<cdiv(totTC, 256), 256, 0, stream>>>(x, w.actA, 16, T0, totTC);

  // 2) down[0] LRU block (H=16, act)
  lru_block<16, true>(d0, w.actA, w.actB, w, BT, stream);

  // 3) down[0] conv: (BT/4 x 64) x (64 x 32) + bias -> skip1
  prep_down_w<<<1, 256, 0, stream>>>(cw0, w.Wconv, 16, 32);
  gemm_wmma<64, 32, true><<<(BT / 4) / 128, 256, 0, stream>>>(w.actB, w.Wconv, cb0,
                                                              w.skip1, BT / 4);

  // 4) down[1] LRU block (H=32, act)
  lru_block<32, true>(d1, w.skip1, w.actB, w, BT / 4, stream);

  // 5) down[1] conv: (BT/16 x 128) x (128 x 64) + bias -> actC
  prep_down_w<<<1, 256, 0, stream>>>(cw1, w.Wconv, 32, 64);
  gemm_wmma<128, 64, true><<<(BT / 16) / 128, 256, 0, stream>>>(w.actB, w.Wconv, cb1,
                                                                w.actC, BT / 16);

  // 6) hidden blocks (H=64, act)
  lru_block<64, true>(h0, w.actC, w.actB, w, BT / 16, stream);
  lru_block<64, true>(h1, w.actB, w.actC, w, BT / 16, stream);

  // 7) up[0]: transpose-conv (64 -> 32, r=4): (BT/16 x 64) x (64 x 128)
  prep_up_w<<<1, 256, 0, stream>>>(uw0, w.Wconv, 64, 32);
  gemm_wmma<64, 128, false><<<(BT / 16) / 128, 256, 0, stream>>>(w.actC, w.Wconv,
                                                                 nullptr, w.bu, BT / 16);
  {
    const int tot = B * (T0 / 4) * 32;  // = BT/4 * 32
    upfix<<<cdiv(tot, 256), 256, 0, stream>>>(w.bu, ub0, w.skip1, w.actB,
                                              T0 / 4, 32, tot);
  }
  lru_block<32, true>(u0, w.actB, w.actC, w, BT / 4, stream);

  // 8) up[1]: transpose-conv (32 -> 16, r=4): (BT/4 x 32) x (32 x 64)
  prep_up_w<<<1, 256, 0, stream>>>(uw1, w.Wconv, 32, 16);
  gemm_wmma<32, 64, false><<<(BT / 4) / 128, 256, 0, stream>>>(w.actC, w.Wconv,
                                                               nullptr, w.bu, BT / 4);
  {
    const int tot = B * T0 * 16;
    upfix<<<cdiv(tot, 256), 256, 0, stream>>>(w.bu, ub1, w.actA, w.actB,
                                              T0, 16, tot);
  }
  lru_block<16, true>(u1, w.actB, w.actA, w, BT, stream);

  // 9) last blocks
  lru_block<16, true>(l0, w.actA, w.actB, w, BT, stream);
  lru_block<16, false>(l1, w.actB, w.actA, w, BT, stream);

  // 10) (B,T,16) -> (B,16,T) into d_out
  tr_tc_ct<<<cdiv(totTC, 256), 256, 0, stream>>>(w.actA, (float*)d_out, 16, T0,
                                                 totTC);
}